// Attention_83949430767970
// MI455X (gfx1250) — compile-verified
//
#include <hip/hip_runtime.h>

#define DK 1024   // hidden
#define SK 8192   // seq len
#define BK 16     // batch
#define NCH 64    // chunks per batch (one block each)
#define SCH (SK / NCH)   // 128 rows per chunk; 16 rows per wave

typedef float v2f __attribute__((ext_vector_type(2)));
typedef float v8f __attribute__((ext_vector_type(8)));

// ---------------------------------------------------------------------------
// Out[16 x 1024] = A[16 x 1024] @ (TRANS ? W^T : W),  W is [1024 x 1024]
// One wave computes one 16x16 tile via 256 chained V_WMMA_F32_16X16X4_F32.
// ---------------------------------------------------------------------------
template <bool TRANS>
__global__ void gemm16_wmma(const float* __restrict__ A,
                            const float* __restrict__ W,
                            float* __restrict__ Out) {
  const int lane  = threadIdx.x & 31;
  const int wave  = threadIdx.x >> 5;
  const int half  = lane >> 4;     // 0: K pair {0,1}, 1: K pair {2,3}
  const int idx16 = lane & 15;     // M for A, N for B
  const int n0    = (blockIdx.x * 8 + wave) * 16;

  const float* Arow = A + idx16 * DK + 2 * half;

  v8f acc = {};
#pragma unroll 4
  for (int k0 = 0; k0 < DK; k0 += 4) {
    const float2 at = *(const float2*)(Arow + k0);
    v2f a; a.x = at.x; a.y = at.y;

    v2f b;
    if (TRANS) {
      const float2 bt = *(const float2*)(W + (n0 + idx16) * DK + k0 + 2 * half);
      b.x = bt.x; b.y = bt.y;
    } else {
      b.x = W[(k0 + 2 * half + 0) * DK + n0 + idx16];
      b.y = W[(k0 + 2 * half + 1) * DK + n0 + idx16];
    }

    acc = __builtin_amdgcn_wmma_f32_16x16x4_f32(
        false, a, false, b, (short)0, acc, false, false);
  }

#pragma unroll
  for (int r = 0; r < 8; ++r)
    Out[(r + 8 * half) * DK + n0 + idx16] = acc[r];
}

// ---------------------------------------------------------------------------
// Fused single-pass: scores + online softmax + weighted encoder accumulation.
// One block per (chunk, b); wave w handles rows s0 + w + 8*i, i = 0..15.
// Emits per-chunk partial (m, l, acc[DK]) and raw scores (for attn output).
// ---------------------------------------------------------------------------
__global__ void fused_pass_kernel(const float* __restrict__ enc,
                                  const float* __restrict__ qt,
                                  float* __restrict__ raw_scores,   // [B,S]
                                  float* __restrict__ pm,           // [B,NCH]
                                  float* __restrict__ pl,           // [B,NCH]
                                  float* __restrict__ pacc) {       // [B,NCH,DK]
  const int b     = blockIdx.y;
  const int chunk = blockIdx.x;
  const int wave  = threadIdx.x >> 5;
  const int lane  = threadIdx.x & 31;
  const int t     = threadIdx.x;
  const int s0    = chunk * SCH;

  __shared__ float  lsm[8];
  __shared__ float  lsl[8];
  __shared__ float4 lacc[8][DK / 4];   // 32 KB

  const float4* q4 = (const float4*)(qt + b * DK);
  float4 qv[8];
#pragma unroll
  for (int k = 0; k < 8; ++k) qv[k] = q4[k * 32 + lane];

  float  m = -3.402823466e38f;
  float  l = 0.f;
  float4 acc[8];
#pragma unroll
  for (int k = 0; k < 8; ++k) acc[k] = make_float4(0.f, 0.f, 0.f, 0.f);

  const size_t enc_b = (size_t)b * SK * DK;
  for (int i = 0; i < SCH / 8; ++i) {
    const int s = s0 + wave + 8 * i;
    const float4* e4 = (const float4*)(enc + enc_b + (size_t)s * DK);
    if (i + 1 < SCH / 8)
      __builtin_prefetch((const void*)(e4 + 8 * (DK / 4) + lane), 0, 0);

    float4 e[8];
#pragma unroll
    for (int k = 0; k < 8; ++k) e[k] = e4[k * 32 + lane];

    float sum = 0.f;
#pragma unroll
    for (int k = 0; k < 8; ++k)
      sum += e[k].x * qv[k].x + e[k].y * qv[k].y + e[k].z * qv[k].z + e[k].w * qv[k].w;
#pragma unroll
    for (int off = 16; off > 0; off >>= 1)
      sum += __shfl_xor(sum, off, 32);          // all lanes get the dot
    const float score = sum * 0.03125f;          // 1/sqrt(1024)
    if (lane == 0) raw_scores[b * SK + s] = score;

    const float mn = fmaxf(m, score);
    const float cf = __expf(m - mn);
    const float p  = __expf(score - mn);
#pragma unroll
    for (int k = 0; k < 8; ++k) {
      acc[k].x = acc[k].x * cf + p * e[k].x;
      acc[k].y = acc[k].y * cf + p * e[k].y;
      acc[k].z = acc[k].z * cf + p * e[k].z;
      acc[k].w = acc[k].w * cf + p * e[k].w;
    }
    l = l * cf + p;
    m = mn;
  }

  // -------- merge the 8 wave partials through LDS (fixed order) --------
  if (lane == 0) { lsm[wave] = m; lsl[wave] = l; }
  __syncthreads();
  float M = lsm[0];
#pragma unroll
  for (int w = 1; w < 8; ++w) M = fmaxf(M, lsm[w]);
  const float sc = __expf(m - M);
#pragma unroll
  for (int k = 0; k < 8; ++k) {
    float4 v = acc[k];
    v.x *= sc; v.y *= sc; v.z *= sc; v.w *= sc;
    lacc[wave][k * 32 + lane] = v;
  }
  __syncthreads();

  float4 r = make_float4(0.f, 0.f, 0.f, 0.f);
#pragma unroll
  for (int w = 0; w < 8; ++w) {
    const float4 v = lacc[w][t];
    r.x += v.x; r.y += v.y; r.z += v.z; r.w += v.w;
  }
  *(float4*)(pacc + ((size_t)(b * NCH + chunk)) * DK + t * 4) = r;

  if (t == 0) {
    float L = 0.f;
#pragma unroll
    for (int w = 0; w < 8; ++w) L += lsl[w] * __expf(lsm[w] - M);
    pm[b * NCH + chunk] = M;
    pl[b * NCH + chunk] = L;
  }
}

// ---------------------------------------------------------------------------
// Per-batch merge of NCH chunk partials -> ctx_raw (already / L), gm, gl.
// ---------------------------------------------------------------------------
__global__ void combine_kernel(const float* __restrict__ pm,
                               const float* __restrict__ pl,
                               const float* __restrict__ pacc,
                               float* __restrict__ ctx_raw,
                               float* __restrict__ gm,
                               float* __restrict__ gl) {
  const int b = blockIdx.x;
  const int t = threadIdx.x;   // 256
  __shared__ float sm[NCH];
  __shared__ float ef[NCH];

  if (t < NCH) sm[t] = pm[b * NCH + t];
  __syncthreads();
  float M = sm[0];
#pragma unroll
  for (int c = 1; c < NCH; ++c) M = fmaxf(M, sm[c]);
  if (t < NCH) ef[t] = __expf(sm[t] - M);
  __syncthreads();

  float L = 0.f;
  for (int c = 0; c < NCH; ++c) L += pl[b * NCH + c] * ef[c];
  const float invL = 1.0f / L;

  float4 r = make_float4(0.f, 0.f, 0.f, 0.f);
  for (int c = 0; c < NCH; ++c) {
    const float4 a = *(const float4*)(pacc + ((size_t)(b * NCH + c)) * DK + t * 4);
    const float  e = ef[c];
    r.x += e * a.x; r.y += e * a.y; r.z += e * a.z; r.w += e * a.w;
  }
  r.x *= invL; r.y *= invL; r.z *= invL; r.w *= invL;
  *(float4*)(ctx_raw + b * DK + t * 4) = r;

  if (t == 0) { gm[b] = M; gl[b] = L; }
}

// ---------------------------------------------------------------------------
// attn[b,s] = exp(raw - gm[b]) / gl[b]   (in place on d_out attn region)
// ---------------------------------------------------------------------------
__global__ void attn_finalize_kernel(float* __restrict__ attn,
                                     const float* __restrict__ gm,
                                     const float* __restrict__ gl) {
  const int i = blockIdx.x * 256 + threadIdx.x;   // 0 .. B*SK-1
  const int b = i >> 13;                          // SK = 8192
  attn[i] = __expf(attn[i] - gm[b]) / gl[b];
}

// ---------------------------------------------------------------------------
extern "C" void kernel_launch(void* const* d_in, const int* in_sizes, int n_in,
                              void* d_out, int out_size, void* d_ws, size_t ws_size,
                              hipStream_t stream) {
  (void)in_sizes; (void)n_in; (void)out_size; (void)ws_size;

  const float* enc = (const float*)d_in[0];  // [B,S,D]
  const float* dec = (const float*)d_in[1];  // [B,D]
  const float* WQ  = (const float*)d_in[2];  // [D,D]
  const float* WK  = (const float*)d_in[3];  // [D,D]
  const float* WV  = (const float*)d_in[4];  // [D,D]

  float* out      = (float*)d_out;
  float* ctx_out  = out;                 // [B,D]   (first return)
  float* attn_out = out + BK * DK;       // [B,1,S] (second return)

  float* ws      = (float*)d_ws;
  float* Q       = ws;                       // 16K floats
  float* qt      = ws + 16384;               // 16K
  float* ctx_raw = ws + 32768;               // 16K
  float* gm      = ws + 49152;               // 16
  float* gl      = ws + 49216;               // 16
  float* pm      = ws + 49280;               // B*NCH = 1024
  float* pl      = ws + 50304;               // 1024
  float* pacc    = ws + 65536;               // B*NCH*DK = 1M floats (4 MB)

  // q~ = (dec @ W_Q^T) @ W_K   (tiny WMMA GEMMs)
  gemm16_wmma<true ><<<8, 256, 0, stream>>>(dec, WQ, Q);
  gemm16_wmma<false><<<8, 256, 0, stream>>>(Q,   WK, qt);

  // single streaming pass over encoder (512 MB): scores + online softmax + acc
  fused_pass_kernel<<<dim3(NCH, BK), 256, 0, stream>>>(enc, qt, attn_out,
                                                       pm, pl, pacc);

  // merge chunk partials; produce ctx_raw (/L) and global (m, l)
  combine_kernel<<<BK, 256, 0, stream>>>(pm, pl, pacc, ctx_raw, gm, gl);

  // finalize attention weights in d_out
  attn_finalize_kernel<<<BK * SK / 256, 256, 0, stream>>>(attn_out, gm, gl);

  // context = ctx_raw @ W_V^T  (tiny WMMA GEMM) -> d_out
  gemm16_wmma<true ><<<8, 256, 0, stream>>>(ctx_raw, WV, ctx_out);
}